// GRANDLayer_11888469475397
// MI455X (gfx1250) — compile-verified
//
#include <hip/hip_runtime.h>
#include <hip/hip_bf16.h>

#define F 32
#define TILE 256          // edges staged per block
#define BLOCK 256         // 8 wave32s

// ---- gfx1250 async-to-LDS path (guarded; falls back to plain LDS stores) ----
#if defined(__gfx1250__) && __has_builtin(__builtin_amdgcn_global_load_async_to_lds_b32)
  #define HAVE_ASYNC_LDS 1
  #if __has_builtin(__builtin_amdgcn_s_wait_asynccnt)
    #define WAIT_ASYNC() __builtin_amdgcn_s_wait_asynccnt(0)
  #else
    #define WAIT_ASYNC() asm volatile("s_wait_asynccnt 0" ::: "memory")
  #endif
#else
  #define HAVE_ASYNC_LDS 0
#endif

// Builtin signature (from hipcc diagnostic): arg0 is global-AS int*, non-const.
typedef __attribute__((address_space(1))) int* gas_i32p;
typedef __attribute__((address_space(3))) int* lds_i32p;

// deg[i] = 1 (self-loop included up front)
__global__ __launch_bounds__(256) void k_init_deg(int* __restrict__ deg, int n) {
    int i = blockIdx.x * blockDim.x + threadIdx.x;
    if (i < n) deg[i] = 1;
}

// in-degree over dst
__global__ __launch_bounds__(256) void k_count_deg(const int* __restrict__ dst,
                                                   int e, int* __restrict__ deg) {
    int i = blockIdx.x * blockDim.x + threadIdx.x;
    if (i < e) atomicAdd(&deg[dst[i]], 1);
}

// dis[i] = deg^-1/2  (deg >= 1 always because of self-loops)
__global__ __launch_bounds__(256) void k_dis(const int* __restrict__ deg,
                                             float* __restrict__ dis, int n) {
    int i = blockIdx.x * blockDim.x + threadIdx.x;
    if (i < n) dis[i] = rsqrtf((float)deg[i]);
}

// self-loop term initializes out: out[i,:] = x[i,:] * dis[i]^2
__global__ __launch_bounds__(256) void k_self(const float* __restrict__ x,
                                              const float* __restrict__ dis,
                                              float* __restrict__ out, int total) {
    int i = blockIdx.x * blockDim.x + threadIdx.x;
    if (i < total) {
        float d = dis[i >> 5];       // F == 32 -> node = i/32
        out[i] = x[i] * d * d;
    }
}

// edge scatter: stage (src,dst) tile into LDS asynchronously, then
// one wave per edge, one lane per feature; atomic f32 scatter-add.
__global__ __launch_bounds__(BLOCK) void k_scatter(const float* __restrict__ x,
                                                   const int* __restrict__ srcp,
                                                   const int* __restrict__ dstp,
                                                   const float* __restrict__ dis,
                                                   float* __restrict__ out, int e) {
    __shared__ int s_src[TILE];
    __shared__ int s_dst[TILE];

    const int tid   = threadIdx.x;
    const int tile0 = blockIdx.x * TILE;
    const int eidx  = tile0 + tid;

#if HAVE_ASYNC_LDS
    if (eidx < e) {
        __builtin_amdgcn_global_load_async_to_lds_b32(
            (gas_i32p)(srcp + eidx), (lds_i32p)&s_src[tid], 0, 0);
        __builtin_amdgcn_global_load_async_to_lds_b32(
            (gas_i32p)(dstp + eidx), (lds_i32p)&s_dst[tid], 0, 0);
    }
    WAIT_ASYNC();
#else
    if (eidx < e) {
        s_src[tid] = srcp[eidx];
        s_dst[tid] = dstp[eidx];
    }
#endif
    __syncthreads();

    const int lane = tid & 31;
    const int base = (tid >> 5) * 32;     // this wave's 32-edge slice of the tile

    #pragma unroll 4
    for (int k = 0; k < 32; ++k) {
        const int idx = base + k;
        const int ge  = tile0 + idx;
        if (ge >= e) break;

        // prefetch the x-row of an edge 2 ahead (gfx1250 global_prefetch_b8)
        const int pn = idx + 2;
        if (k + 2 < 32 && tile0 + pn < e) {
            const int sp = s_src[pn];
            __builtin_prefetch((const void*)&x[sp * F + lane], 0, 3);
        }

        const int s = s_src[idx];          // wave-uniform LDS broadcast
        const int d = s_dst[idx];
        const float w = dis[s] * dis[d];
        const float v = x[s * F + lane] * w;   // coalesced 128B row read (L2-resident)
        atomicAdd(&out[d * F + lane], v);      // no-return global_atomic_add_f32
    }
}

extern "C" void kernel_launch(void* const* d_in, const int* in_sizes, int n_in,
                              void* d_out, int out_size, void* d_ws, size_t ws_size,
                              hipStream_t stream) {
    const float* x  = (const float*)d_in[0];
    const int*   ei = (const int*)d_in[1];
    const int n = in_sizes[0] / F;     // 100000
    const int e = in_sizes[1] / 2;     // 1600000
    const int* src = ei;               // edge_index[0,:]
    const int* dst = ei + e;           // edge_index[1,:]
    float* out = (float*)d_out;

    int*   deg = (int*)d_ws;                   // n ints
    float* dis = (float*)((int*)d_ws + n);     // n floats

    const int nb_n  = (n + 255) / 256;
    const int nb_e  = (e + 255) / 256;
    const int total = n * F;
    const int nb_nf = (total + 255) / 256;
    const int nb_sc = (e + TILE - 1) / TILE;

    k_init_deg <<<nb_n,  256, 0, stream>>>(deg, n);
    k_count_deg<<<nb_e,  256, 0, stream>>>(dst, e, deg);
    k_dis      <<<nb_n,  256, 0, stream>>>(deg, dis, n);
    k_self     <<<nb_nf, 256, 0, stream>>>(x, dis, out, total);
    k_scatter  <<<nb_sc, BLOCK, 0, stream>>>(x, src, dst, dis, out, e);
}